// MACD_73701638799847
// MI455X (gfx1250) — compile-verified
//
#include <hip/hip_runtime.h>

// MACD via WMMA blocked scan on gfx1250 (wave32).
// y_t = a*x_t + (1-a)*y_{t-1} over 16-step blocks:
//   Y = Ltri * X + pow * carry,  Ltri[m,k] = a*(1-a)^(m-k) (k<=m)
// D = A(16x16 tri, f32) x B(16 times x 16 rows) via 4 chained v_wmma_f32_16x16x4_f32.
// Chunked over T with warmup so all waves are independent (decay^1024 << fp32 eps).

typedef float v2f __attribute__((ext_vector_type(2)));
typedef float v4f __attribute__((ext_vector_type(4)));
typedef float v8f __attribute__((ext_vector_type(8)));

#define B_ROWS 512
#define T_LEN  16384
#define CHUNK  512      // output timesteps per workgroup
#define WARM   1024     // warmup timesteps ((1-a_long)^1024 ~ 1e-35)
#define RG     16       // rows per workgroup (one 16x16 tile)
#define PF_AHEAD 256    // prefetch distance in floats (1 KiB/row, ~16 iters ahead)

// Broadcast y[M=15, N=row] (held in lanes 16..31 of the VGPR) to every lane's
// matching row = lane&15, via ds_bpermute_b32 (byte indices).
__device__ __forceinline__ float bcast_carry(float v) {
  int idx = (((int)(threadIdx.x & 15) + 16) << 2);
  return __int_as_float(__builtin_amdgcn_ds_bpermute(idx, __float_as_int(v)));
}

// Exact integer power via branchless binary exponentiation (e in [0,31]).
// Replaces __powf: no transcendental lib code, ~10 VALU ops, uniform control flow.
__device__ __forceinline__ float upow(float base, int e) {
  float w = 1.0f;
  float sq = base;
#pragma unroll
  for (int b = 0; b < 5; ++b) {
    w *= (e & (1 << b)) ? sq : 1.0f;   // v_cndmask + v_mul
    sq *= sq;
  }
  return w;
}

struct Chain {
  v2f   a[4];   // this lane's A-matrix values per K-slice (A 16x4 layout)
  float p[8];   // (1-a)^(m+1) for m = v + 8*half  (D layout)
  float c;      // per-lane broadcast carry: carry[row = lane&15]
};

__device__ __forceinline__ void chain_init(Chain& ch, float alpha, int n, int half) {
  const float oma = 1.0f - alpha;
#pragma unroll
  for (int ks = 0; ks < 4; ++ks) {
#pragma unroll
    for (int j = 0; j < 2; ++j) {
      // A 16x4 f32 layout: lane holds M = lane&15; VGPR j, lane-half => K = ks*4 + 2*half + j
      int k = ks * 4 + half * 2 + j;
      ch.a[ks][j] = (k <= n) ? alpha * upow(oma, n - k) : 0.0f;
    }
  }
#pragma unroll
  for (int v = 0; v < 8; ++v) {
    int m = v + 8 * half;                 // D layout: M = vgpr + 8*(lane>=16)
    ch.p[v] = upow(oma, m + 1);
  }
  ch.c = 0.0f;
}

// One 16-timestep EMA block-step: 4 chained f32 WMMAs + carry fixup + carry extract.
__device__ __forceinline__ v8f chain_step(Chain& ch, const v2f xb[4]) {
  v8f d = {};
#pragma unroll
  for (int ks = 0; ks < 4; ++ks) {
    d = __builtin_amdgcn_wmma_f32_16x16x4_f32(
        /*neg_a=*/false, ch.a[ks], /*neg_b=*/false, xb[ks],
        /*c_mod=*/(short)0, d, /*reuse_a=*/false, /*reuse_b=*/false);
  }
  v8f y;
#pragma unroll
  for (int v = 0; v < 8; ++v) y[v] = d[v] + ch.p[v] * ch.c;
  ch.c = bcast_carry(y[7]);   // y[M=15, row] lives in VGPR7, lanes 16..31
  return y;
}

__global__ void macd_wmma_kernel(const float* __restrict__ x,
                                 const int* __restrict__ p_short,
                                 const int* __restrict__ p_long,
                                 const int* __restrict__ p_sig,
                                 float* __restrict__ out) {
  __shared__ float lds[256];              // 16x16 macd tile for D->B relayout

  const int lane = threadIdx.x;
  const int half = lane >> 4;             // 0: lanes 0-15, 1: lanes 16-31
  const int n    = lane & 15;             // tile column: row index for A/B/D "N"

  const int rowgroup = blockIdx.x & (B_ROWS / RG - 1);
  const int chunk    = blockIdx.x / (B_ROWS / RG);
  const int row      = rowgroup * RG + n;

  const float as = 2.0f / (float)(*p_short + 1);
  const float al = 2.0f / (float)(*p_long + 1);
  const float ag = 2.0f / (float)(*p_sig + 1);

  Chain cs, cl, cg;
  chain_init(cs, as, n, half);
  chain_init(cl, al, n, half);
  chain_init(cg, ag, n, half);

  const int t_out = chunk * CHUNK;
  int t_begin = t_out - WARM;
  if (t_begin < 0) t_begin = 0;
  const int t_end = t_out + CHUNK;

  const float* xrow = x + (size_t)row * T_LEN;
  if (t_begin == 0) {
    // Exact init: y_0 = x_0  <=>  carry = x_0 (macd_0 == 0 => signal carry 0).
    float x0 = xrow[0];
    cs.c = x0;
    cl.c = x0;
    cg.c = 0.0f;
  }

  float* om = out;                                    // macd line
  float* os = out + (size_t)B_ROWS * T_LEN;           // signal line
  float* oh = out + (size_t)2 * B_ROWS * T_LEN;       // histogram

  for (int tb = t_begin; tb < t_end; tb += 16) {
    // Speculative prefetch of this row's stream, ~16 iterations ahead.
    {
      int tp = tb + PF_AHEAD;
      if (tp > T_LEN - 16) tp = T_LEN - 16;           // uniform clamp
      __builtin_prefetch(xrow + tp, 0, 0);            // global_prefetch_b8
    }

    // B operand (4x16 f32 per slice): lane holds N=row=n, K = ks*4 + 2*half + j
    v2f xb[4];
#pragma unroll
    for (int ks = 0; ks < 4; ++ks)
      xb[ks] = *(const v2f*)(xrow + tb + ks * 4 + half * 2);

    v8f ys = chain_step(cs, xb);
    v8f yl = chain_step(cl, xb);

    float md[8];
#pragma unroll
    for (int v = 0; v < 8; ++v) md[v] = ys[v] - yl[v];

    // Relayout macd tile: D layout [m = v+8*half, n] -> B layout for signal chain.
#pragma unroll
    for (int v = 0; v < 8; ++v) lds[(v + 8 * half) * 16 + n] = md[v];
    __syncthreads();
    v2f mb[4];
#pragma unroll
    for (int ks = 0; ks < 4; ++ks) {
      int k = ks * 4 + half * 2;
      mb[ks][0] = lds[k * 16 + n];
      mb[ks][1] = lds[(k + 1) * 16 + n];
    }
    __syncthreads();

    v8f yg = chain_step(cg, mb);

    if (tb >= t_out) {
      // Lane's 8 D values are 8 consecutive timesteps: t = tb + 8*half + v.
      size_t base = (size_t)row * T_LEN + (size_t)(tb + 8 * half);
      v4f m0 = {md[0], md[1], md[2], md[3]};
      v4f m1 = {md[4], md[5], md[6], md[7]};
      v4f s0 = {yg[0], yg[1], yg[2], yg[3]};
      v4f s1 = {yg[4], yg[5], yg[6], yg[7]};
      v4f h0 = {md[0] - yg[0], md[1] - yg[1], md[2] - yg[2], md[3] - yg[3]};
      v4f h1 = {md[4] - yg[4], md[5] - yg[5], md[6] - yg[6], md[7] - yg[7]};
      *(v4f*)(om + base)     = m0;
      *(v4f*)(om + base + 4) = m1;
      *(v4f*)(os + base)     = s0;
      *(v4f*)(os + base + 4) = s1;
      *(v4f*)(oh + base)     = h0;
      *(v4f*)(oh + base + 4) = h1;
    }
  }
}

extern "C" void kernel_launch(void* const* d_in, const int* in_sizes, int n_in,
                              void* d_out, int out_size, void* d_ws, size_t ws_size,
                              hipStream_t stream) {
  (void)in_sizes; (void)n_in; (void)out_size; (void)d_ws; (void)ws_size;
  const float* x  = (const float*)d_in[0];
  const int*   ps = (const int*)d_in[1];
  const int*   pl = (const int*)d_in[2];
  const int*   pg = (const int*)d_in[3];
  float*       o  = (float*)d_out;

  dim3 grid((B_ROWS / RG) * (T_LEN / CHUNK));   // 32 row-groups x 32 chunks = 1024 WGs
  dim3 block(32);                               // one wave32 per workgroup
  macd_wmma_kernel<<<grid, block, 0, stream>>>(x, ps, pl, pg, o);
}